// TargetDrop_36009005810158
// MI455X (gfx1250) — compile-verified
//
#include <hip/hip_runtime.h>

#define B_    64
#define C_    256
#define H_    56
#define W_    56
#define HW_   3136
#define HW4_  784
#define D_    16
#define BC_   16384
#define TOPK_ 38

typedef __attribute__((ext_vector_type(2))) float v2f;
typedef __attribute__((ext_vector_type(8))) float v8f;

// ---------------------------------------------------------------------------
// Kernel 1: per-(b,c) row: sum (for avg-pool) and first-occurrence argmax.
// One block per row, float4 (b128) loads, LDS tree reduction.
// ---------------------------------------------------------------------------
__global__ __launch_bounds__(256) void td_reduce(const float* __restrict__ x,
                                                 float* __restrict__ sumg,
                                                 int* __restrict__ idxg) {
  __shared__ float ssum[256];
  __shared__ float smax[256];
  __shared__ int   sidx[256];
  const int r   = blockIdx.x;
  const int tid = threadIdx.x;
  const float4* xr = (const float4*)x + (size_t)r * HW4_;

  float acc  = 0.0f;
  float bmax = -3.4e38f;
  int   bidx = 0;
  for (int i = tid; i < HW4_; i += 256) {
    float4 v = xr[i];
    acc += (v.x + v.y) + (v.z + v.w);
    int base = i * 4;
    if (v.x > bmax) { bmax = v.x; bidx = base;     }
    if (v.y > bmax) { bmax = v.y; bidx = base + 1; }
    if (v.z > bmax) { bmax = v.z; bidx = base + 2; }
    if (v.w > bmax) { bmax = v.w; bidx = base + 3; }
  }
  ssum[tid] = acc; smax[tid] = bmax; sidx[tid] = bidx;
  __syncthreads();
  for (int s = 128; s > 0; s >>= 1) {
    if (tid < s) {
      ssum[tid] += ssum[tid + s];
      float ov = smax[tid + s]; int oi = sidx[tid + s];
      if (ov > smax[tid] || (ov == smax[tid] && oi < sidx[tid])) {
        smax[tid] = ov; sidx[tid] = oi;
      }
    }
    __syncthreads();
  }
  if (tid == 0) { sumg[r] = ssum[0]; idxg[r] = sidx[0]; }
}

// ---------------------------------------------------------------------------
// Kernel 2: SE MLP via V_WMMA_F32_16X16X4_F32 + sigmoid + top-k rank +
// drop-box bounds / lambda. Single 1024-thread block (32 wave32 waves).
// In-place ws reuse: sumg_in -> lamg, idxg -> boundsg (no restrict on those).
// ---------------------------------------------------------------------------
__global__ __launch_bounds__(1024) void td_se(const float* sumg_in,
                                              const int* idxg,
                                              const float* __restrict__ w1,
                                              const float* __restrict__ w2,
                                              int* boundsg,
                                              float* lamg) {
  extern __shared__ char smem[];
  float* hidden = (float*)smem;                         // 64 x 16 f32
  float* Mm     = (float*)(smem + (size_t)64 * D_ * 4); // 64 x 256 f32

  const int tid  = threadIdx.x;
  const int wv   = tid >> 5;
  const int lane = tid & 31;
  const int hi   = (lane >> 4) & 1;   // upper half-wave holds K+2/K+3
  const int l16  = lane & 15;
  const float invHW = 1.0f / (float)HW_;

  // ---- phase 1: hidden = relu(pooled @ w1^T), 4 M-tiles on waves 0..3 ----
  if (wv < 4) {
    const int m0 = wv * 16;
    const int mr = m0 + l16;
    v8f acc = {};
    for (int k0 = 0; k0 < C_; k0 += 4) {
      const int koff = k0 + (hi ? 2 : 0);
      float2 av = *(const float2*)(sumg_in + mr * C_ + koff);  // pooled = sum/HW
      float2 bv = *(const float2*)(w1 + l16 * C_ + koff);      // B[k][n] = w1[n][k]
      v2f a; a.x = av.x * invHW; a.y = av.y * invHW;
      v2f b; b.x = bv.x;         b.y = bv.y;
      acc = __builtin_amdgcn_wmma_f32_16x16x4_f32(false, a, false, b,
                                                  (short)0, acc, false, false);
    }
    const int mrow = m0 + (hi ? 8 : 0);
#pragma unroll
    for (int v = 0; v < 8; ++v)
      hidden[(mrow + v) * D_ + l16] = fmaxf(acc[v], 0.0f);
  }
  __syncthreads();

  // ---- phase 2: M = sigmoid(hidden @ w2^T), 64 tiles, 2 per wave ----
#pragma unroll
  for (int t2 = 0; t2 < 2; ++t2) {
    const int t  = wv * 2 + t2;
    const int m0 = (t >> 4) * 16;
    const int n0 = (t & 15) * 16;
    v8f acc = {};
#pragma unroll
    for (int k0 = 0; k0 < D_; k0 += 4) {
      const int koff = k0 + (hi ? 2 : 0);
      float2 av = *(const float2*)(hidden + (m0 + l16) * D_ + koff);
      float2 bv = *(const float2*)(w2 + (n0 + l16) * D_ + koff);  // B[k][n]=w2[n][k]
      v2f a; a.x = av.x; a.y = av.y;
      v2f b; b.x = bv.x; b.y = bv.y;
      acc = __builtin_amdgcn_wmma_f32_16x16x4_f32(false, a, false, b,
                                                  (short)0, acc, false, false);
    }
    const int mrow = m0 + (hi ? 8 : 0);
#pragma unroll
    for (int v = 0; v < 8; ++v)
      Mm[(mrow + v) * C_ + n0 + l16] = 1.0f / (1.0f + expf(-acc[v]));
  }
  __syncthreads();

  // ---- phase 3: exact top-38 rank per batch row + bounds/lambda ----
#pragma unroll
  for (int rr = 0; rr < 2; ++rr) {
    const int b = wv * 2 + rr;          // 32 waves x 2 rows = 64 rows
    const float* row = Mm + b * C_;
    float myv[8]; int rank[8];
#pragma unroll
    for (int j = 0; j < 8; ++j) { myv[j] = row[lane * 8 + j]; rank[j] = 0; }
    for (int j2 = 0; j2 < C_; ++j2) {
      float mv = row[j2];               // LDS broadcast read (conflict-free)
#pragma unroll
      for (int j = 0; j < 8; ++j) {
        int cj = lane * 8 + j;          // tie-break: earlier index wins
        rank[j] += (int)((mv > myv[j]) || ((mv == myv[j]) && (j2 < cj)));
      }
    }
#pragma unroll
    for (int j = 0; j < 8; ++j) {
      int c = lane * 8 + j;
      int r = b * C_ + c;
      bool flag = rank[j] < TOPK_;
      int mi = idxg[r];
      int mh = mi / W_, mw = mi - mh * W_;
      int h1 = mh - 2; if (h1 < 0) h1 = 0;
      int h2 = mh + 2; if (h2 > H_ - 1) h2 = H_ - 1;
      int wl = mw - 2; if (wl < 0) wl = 0;
      int wr = mw + 2; if (wr > W_ - 1) wr = W_ - 1;
      int area = (h2 - h1 + 1) * (wr - wl + 1);
      float lam = (float)HW_ / (float)(HW_ - area);
      // unselected rows: impossible box (h1=255) and lambda=1 -> out == x
      boundsg[r] = flag ? (h1 | (h2 << 8) | (wl << 16) | (wr << 24)) : 255;
      lamg[r]    = flag ? lam : 1.0f;
    }
  }
}

// ---------------------------------------------------------------------------
// Kernel 3: apply mask: out = in_box ? 0 : x * lambda. float4 read+write.
// ---------------------------------------------------------------------------
__global__ __launch_bounds__(256) void td_apply(const float* __restrict__ x,
                                                const int* __restrict__ boundsg,
                                                const float* __restrict__ lamg,
                                                float* __restrict__ out) {
  const int r   = blockIdx.x;
  const int tid = threadIdx.x;
  const int pb    = boundsg[r];
  const float lam = lamg[r];
  const int h1 = pb & 255, h2 = (pb >> 8) & 255;
  const int wl = (pb >> 16) & 255, wr = (pb >> 24) & 255;
  const float4* xr = (const float4*)x + (size_t)r * HW4_;
  float4* orr      = (float4*)out + (size_t)r * HW4_;
  for (int i = tid; i < HW4_; i += 256) {
    float4 v = xr[i];
    int base = i * 4;
    float o[4] = {v.x, v.y, v.z, v.w};
#pragma unroll
    for (int j = 0; j < 4; ++j) {
      int idx = base + j;
      int h = idx / W_;
      int w = idx - h * W_;
      bool ib = (h >= h1) & (h <= h2) & (w >= wl) & (w <= wr);
      o[j] = ib ? 0.0f : o[j] * lam;
    }
    float4 ov; ov.x = o[0]; ov.y = o[1]; ov.z = o[2]; ov.w = o[3];
    orr[i] = ov;
  }
}

extern "C" void kernel_launch(void* const* d_in, const int* in_sizes, int n_in,
                              void* d_out, int out_size, void* d_ws, size_t ws_size,
                              hipStream_t stream) {
  (void)in_sizes; (void)n_in; (void)out_size; (void)ws_size;
  const float* x  = (const float*)d_in[0];   // [64,256,56,56] f32
  const float* w1 = (const float*)d_in[1];   // [16,256] f32
  const float* w2 = (const float*)d_in[2];   // [256,16] f32
  float* out = (float*)d_out;

  char* ws = (char*)d_ws;
  float* sum_lam = (float*)ws;                // 16384 f32: row sums -> lambda
  int*   idx_bnd = (int*)(ws + 65536);        // 16384 i32: argmax  -> bounds
  // total workspace use: 128 KB

  td_reduce<<<BC_, 256, 0, stream>>>(x, sum_lam, idx_bnd);

  const size_t smem = ((size_t)64 * D_ + (size_t)64 * C_) * sizeof(float); // 68 KB
  hipFuncSetAttribute((const void*)td_se,
                      hipFuncAttributeMaxDynamicSharedMemorySize, (int)smem);
  td_se<<<1, 1024, smem, stream>>>(sum_lam, idx_bnd, w1, w2, idx_bnd, sum_lam);

  td_apply<<<BC_, 256, 0, stream>>>(x, idx_bnd, sum_lam, out);
}